// advanced_GCN_72679436583080
// MI455X (gfx1250) — compile-verified
//
#include <hip/hip_runtime.h>
#include <hip/hip_bf16.h>

typedef __attribute__((ext_vector_type(2))) float v2f;
typedef __attribute__((ext_vector_type(8))) float v8f;

#define HID 64
#define NFEAT 128
#define BN_EPS 1e-5f

// D(16x16,f32) = A(16x4,f32) x B(4x16,f32) + C  -- full fp32 WMMA on gfx1250
static __device__ __forceinline__ v8f wmma4(v2f a, v2f b, v8f c) {
  return __builtin_amdgcn_wmma_f32_16x16x4_f32(
      /*neg_a=*/false, a, /*neg_b=*/false, b,
      /*c_mod=*/(short)0, c, /*reuse_a=*/false, /*reuse_b=*/false);
}

// O[N,64] = H[N,K] @ W[64,K]^T   (one wave -> 16 rows x all 64 cols)
template <int K>
__global__ void gemm_nodes_wmma(const float* __restrict__ H,
                                const float* __restrict__ W,
                                float* __restrict__ O, int nrows) {
  const int lane = threadIdx.x & 31;
  const int wave = blockIdx.x * (blockDim.x >> 5) + (threadIdx.x >> 5);
  const int m0 = wave * 16;
  if (m0 >= nrows) return;  // wave-uniform: EXEC stays all-1s for live waves

  const int mlane = lane & 15;
  const int khalf = (lane >> 4) << 1;  // 0 or 2: K sub-chunk per lane-half
  int row = m0 + mlane;
  if (row >= nrows) row = nrows - 1;   // clamp loads for (non-existent here) partial tiles

  const float* hrow = H + (size_t)row * K + khalf;
  const float* w0 = W + (size_t)(mlane + 0) * K + khalf;
  const float* w1 = W + (size_t)(mlane + 16) * K + khalf;
  const float* w2 = W + (size_t)(mlane + 32) * K + khalf;
  const float* w3 = W + (size_t)(mlane + 48) * K + khalf;

  v8f c0 = {}, c1 = {}, c2 = {}, c3 = {};
#pragma unroll 4
  for (int kb = 0; kb < K; kb += 4) {
    v2f a  = *(const v2f*)(hrow + kb);
    v2f b0 = *(const v2f*)(w0 + kb);
    v2f b1 = *(const v2f*)(w1 + kb);
    v2f b2 = *(const v2f*)(w2 + kb);
    v2f b3 = *(const v2f*)(w3 + kb);
    c0 = wmma4(a, b0, c0);
    c1 = wmma4(a, b1, c1);
    c2 = wmma4(a, b2, c2);
    c3 = wmma4(a, b3, c3);
  }

  const int rbase = m0 + ((lane >> 4) << 3);  // lanes 16-31 hold M = v+8
#pragma unroll
  for (int v = 0; v < 8; ++v) {
    int orow = rbase + v;
    if (orow < nrows) {
      float* o = O + (size_t)orow * HID + mlane;
      o[0]  = c0[v];
      o[16] = c1[v];
      o[32] = c2[v];
      o[48] = c3[v];
    }
  }
}

__global__ void k_fill(float* __restrict__ p, float v, int n) {
  int i = blockIdx.x * blockDim.x + threadIdx.x;
  if (i < n) p[i] = v;
}

__global__ void k_deg_edges(const int* __restrict__ dst, float* __restrict__ deg, int E) {
  int e = blockIdx.x * blockDim.x + threadIdx.x;
  if (e < E) atomicAdd(&deg[dst[e]], 1.0f);
}

__global__ void k_rsqrt_inplace(float* __restrict__ p, int n) {
  int i = blockIdx.x * blockDim.x + threadIdx.x;
  if (i < n) p[i] = rsqrtf(p[i]);
}

// Out[i,f] = Hp[i,f] * dinv[i]^2  (self-loop term; also fully initializes Out)
__global__ void k_self_init(const float* __restrict__ Hp, const float* __restrict__ dinv,
                            float* __restrict__ Out, int total) {
  int i = blockIdx.x * blockDim.x + threadIdx.x;
  if (i < total) {
    float dv = dinv[i >> 6];
    Out[i] = Hp[i] * dv * dv;
  }
}

// one wave per edge: 64 features = 2 per lane; gather + scatter-add
__global__ void k_agg_edges(const float* __restrict__ Hp, const int* __restrict__ src,
                            const int* __restrict__ dst, const float* __restrict__ dinv,
                            float* __restrict__ Out, int E) {
  int wid = blockIdx.x * (blockDim.x >> 5) + (threadIdx.x >> 5);
  if (wid >= E) return;
  int lane = threadIdx.x & 31;
  int s = src[wid];
  int d = dst[wid];
  float nrm = dinv[s] * dinv[d];
  v2f m = *(const v2f*)(Hp + (size_t)s * HID + lane * 2);
  float* o = Out + (size_t)d * HID + lane * 2;
  atomicAdd(o + 0, m.x * nrm);
  atomicAdd(o + 1, m.y * nrm);
}

// Hcur = relu(bn(Agg + bias)) [+ Hcur if residual]
__global__ void k_post(const float* __restrict__ Agg, const float* __restrict__ bias,
                       const float* __restrict__ gam, const float* __restrict__ bet,
                       const float* __restrict__ mu, const float* __restrict__ var,
                       float* __restrict__ Hcur, int total, int residual) {
  int i = blockIdx.x * blockDim.x + threadIdx.x;
  if (i >= total) return;
  int f = i & (HID - 1);
  float v = Agg[i] + bias[f];
  v = (v - mu[f]) * (gam[f] * rsqrtf(var[f] + BN_EPS)) + bet[f];
  v = fmaxf(v, 0.0f);
  Hcur[i] = residual ? (v + Hcur[i]) : v;
}

// one wave per node: accumulate into per-graph sums + counts
__global__ void k_pool_acc(const float* __restrict__ H, const int* __restrict__ batch,
                           float* __restrict__ pooled, float* __restrict__ cnts, int N) {
  int wid = blockIdx.x * (blockDim.x >> 5) + (threadIdx.x >> 5);
  if (wid >= N) return;
  int lane = threadIdx.x & 31;
  int g = batch[wid];
  v2f h = *(const v2f*)(H + (size_t)wid * HID + lane * 2);
  atomicAdd(pooled + (size_t)g * HID + lane * 2 + 0, h.x);
  atomicAdd(pooled + (size_t)g * HID + lane * 2 + 1, h.y);
  if (lane == 0) atomicAdd(cnts + g, 1.0f);
}

__global__ void k_pool_final(const float* __restrict__ pooled, const float* __restrict__ cnts,
                             const float* __restrict__ lw, const float* __restrict__ lb,
                             float* __restrict__ out, int G) {
  int g = blockIdx.x * blockDim.x + threadIdx.x;
  if (g >= G) return;
  float c = fmaxf(cnts[g], 1.0f);
  float acc = 0.0f;
  for (int f = 0; f < HID; ++f) acc += pooled[(size_t)g * HID + f] * lw[f];
  out[g] = acc / c + lb[0];
}

extern "C" void kernel_launch(void* const* d_in, const int* in_sizes, int n_in,
                              void* d_out, int out_size, void* d_ws, size_t ws_size,
                              hipStream_t stream) {
  const float* x     = (const float*)d_in[0];
  const int*   ei    = (const int*)d_in[1];
  const int*   batch = (const int*)d_in[2];
  const float* W_in  = (const float*)d_in[3];
  const float* W1    = (const float*)d_in[4];
  const float* b1    = (const float*)d_in[5];
  const float* Ws    = (const float*)d_in[6];
  const float* bs    = (const float*)d_in[7];
  const float* bn_g  = (const float*)d_in[8];
  const float* bn_b  = (const float*)d_in[9];
  const float* bn_m  = (const float*)d_in[10];
  const float* bn_v  = (const float*)d_in[11];
  const float* lin_w = (const float*)d_in[12];
  const float* lin_b = (const float*)d_in[13];
  float* out = (float*)d_out;

  const int N = in_sizes[0] / NFEAT;   // 50000
  const int E = in_sizes[1] / 2;       // 800000
  const int G = out_size;              // 64
  const int* src = ei;
  const int* dst = ei + E;

  // workspace layout (floats)
  float* ws     = (float*)d_ws;
  float* dinv   = ws;                        // N      (deg -> rsqrt in place)
  float* Hproj  = dinv + N;                  // N*HID
  float* Hout   = Hproj + (size_t)N * HID;   // N*HID
  float* Hcur   = Hout + (size_t)N * HID;    // N*HID
  float* pooled = Hcur + (size_t)N * HID;    // G*HID
  float* cnts   = pooled + (size_t)G * HID;  // G (contiguous after pooled)

  const int T = 256;
  const int total = N * HID;
  auto b1d = [](int n) { return (n + 255) / 256; };
  const int tilesM = (N + 15) / 16;
  const int gemmBlocks = (tilesM + 7) / 8;      // 8 waves per block
  const int edgeWaveBlocks = (E + 7) / 8;       // 1 wave per edge
  const int nodeWaveBlocks = (N + 7) / 8;       // 1 wave per node

  // degrees -> dinv
  k_fill<<<b1d(N), T, 0, stream>>>(dinv, 1.0f, N);  // self-loop contributes 1
  k_deg_edges<<<b1d(E), T, 0, stream>>>(dst, dinv, E);
  k_rsqrt_inplace<<<b1d(N), T, 0, stream>>>(dinv, N);

  // identity projection -> Hcur ; layer-1 projection -> Hproj
  gemm_nodes_wmma<NFEAT><<<gemmBlocks, T, 0, stream>>>(x, W_in, Hcur, N);
  gemm_nodes_wmma<NFEAT><<<gemmBlocks, T, 0, stream>>>(x, W1, Hproj, N);
  k_self_init<<<b1d(total), T, 0, stream>>>(Hproj, dinv, Hout, total);
  k_agg_edges<<<edgeWaveBlocks, T, 0, stream>>>(Hproj, src, dst, dinv, Hout, E);
  k_post<<<b1d(total), T, 0, stream>>>(Hout, b1, bn_g, bn_b, bn_m, bn_v, Hcur, total, 1);

  // layers 2-4 with residual
  for (int l = 0; l < 3; ++l) {
    gemm_nodes_wmma<HID><<<gemmBlocks, T, 0, stream>>>(Hcur, Ws + (size_t)l * HID * HID, Hproj, N);
    k_self_init<<<b1d(total), T, 0, stream>>>(Hproj, dinv, Hout, total);
    k_agg_edges<<<edgeWaveBlocks, T, 0, stream>>>(Hproj, src, dst, dinv, Hout, E);
    k_post<<<b1d(total), T, 0, stream>>>(Hout, bs + (size_t)l * HID,
                                         bn_g + (size_t)(l + 1) * HID, bn_b + (size_t)(l + 1) * HID,
                                         bn_m + (size_t)(l + 1) * HID, bn_v + (size_t)(l + 1) * HID,
                                         Hcur, total, 1);
  }

  // layer 5, no residual
  gemm_nodes_wmma<HID><<<gemmBlocks, T, 0, stream>>>(Hcur, Ws + (size_t)3 * HID * HID, Hproj, N);
  k_self_init<<<b1d(total), T, 0, stream>>>(Hproj, dinv, Hout, total);
  k_agg_edges<<<edgeWaveBlocks, T, 0, stream>>>(Hproj, src, dst, dinv, Hout, E);
  k_post<<<b1d(total), T, 0, stream>>>(Hout, bs + (size_t)3 * HID,
                                       bn_g + (size_t)4 * HID, bn_b + (size_t)4 * HID,
                                       bn_m + (size_t)4 * HID, bn_v + (size_t)4 * HID,
                                       Hcur, total, 0);

  // global mean pool + final linear
  k_fill<<<b1d(G * HID + G), T, 0, stream>>>(pooled, 0.0f, G * HID + G);  // pooled + cnts
  k_pool_acc<<<nodeWaveBlocks, T, 0, stream>>>(Hcur, batch, pooled, cnts, N);
  k_pool_final<<<1, 64, 0, stream>>>(pooled, cnts, lin_w, lin_b, out, G);
}